// GATLayer_523986010187
// MI455X (gfx1250) — compile-verified
//
#include <hip/hip_runtime.h>
#include <hip/hip_bf16.h>

// Problem constants (match reference)
#define BATCH 512
#define NNODE 128
#define CIN   64
#define COUT  64
#define RREL  4
#define ALPHA 0.2f
#define JR    (NNODE * RREL)   // 512  joint (neighbor, relation) dim
#define HROW  (COUT * RREL)    // 256  projected row width

#define LROWS  516             // padded LDS row stride (floats): banks 4*r, 16B aligned
#define NEGBIG -3.4e38f

typedef __attribute__((ext_vector_type(2))) float v2f;
typedef __attribute__((ext_vector_type(4))) float v4f;
typedef __attribute__((ext_vector_type(8))) float v8f;

// D = A(16x4) * B(4x16) + C, fp32 WMMA (CDNA5)
__device__ __forceinline__ v8f wmma4(v2f a, v2f b, v8f c) {
    return __builtin_amdgcn_wmma_f32_16x16x4_f32(
        /*neg_a=*/false, a, /*neg_b=*/false, b,
        /*c_mod=*/(short)0, c, /*reuse_a=*/false, /*reuse_b=*/false);
}

// ---------------------------------------------------------------------------
// Kernel 1: h[b] = atoms[b] (128x64) @ W (64x256) + bias   (fp32 WMMA)
// grid = 512 (block per batch), block = 256 (8 waves, wave = 16-row M-tile).
// W (64 KB) is staged once per block into LDS; B-fragments come from ds_load,
// so the WMMA stream no longer stalls on strided global loads.
// ---------------------------------------------------------------------------
__global__ __launch_bounds__(256) void gat_proj_kernel(
    const float* __restrict__ atoms, const float* __restrict__ W,
    const float* __restrict__ bias, float* __restrict__ h) {
    __shared__ float sW[CIN * HROW];   // 64 KB

    const int tid  = threadIdx.x;
    const int b    = blockIdx.x;
    const int wave = tid >> 5;
    const int lane = tid & 31;
    const int m    = lane & 15;   // A row / B,D col within tile
    const int half = lane >> 4;   // k-half of the 16x4 fragment
    const int i0   = wave * 16;

    // Cooperative coalesced stage of W: 16384 floats / 256 threads = 16 x v4f
#pragma unroll
    for (int v = 0; v < 16; ++v) {
        const int off = (v * 256 + tid) * 4;          // float index
        *(v4f*)(sW + off) = *(const v4f*)(W + off);
    }

    // Preload this lane's A slots for all 16 K-steps (consecutive float2)
    const float* arow = atoms + ((size_t)b * NNODE + i0 + m) * CIN;
    v2f afrag[16];
#pragma unroll
    for (int kk = 0; kk < 16; ++kk)
        afrag[kk] = *(const v2f*)(arow + kk * 4 + 2 * half);

    __syncthreads();

    float* hb = h + (size_t)b * NNODE * HROW;

    for (int nt = 0; nt < 16; ++nt) {
        const int n0 = nt * 16;
        v8f acc = {};
#pragma unroll
        for (int kk = 0; kk < 16; ++kk) {
            const int kbase = kk * 4 + 2 * half;      // k of vgpr slot v=0
            v2f bfrag;
            bfrag.x = sW[(kbase + 0) * HROW + n0 + m];
            bfrag.y = sW[(kbase + 1) * HROW + n0 + m];
            acc = wmma4(afrag[kk], bfrag, acc);
        }
        const float bv = bias[n0 + m];
#pragma unroll
        for (int v = 0; v < 8; ++v) {
            const int row = i0 + v + 8 * half;        // D layout
            hb[(size_t)row * HROW + n0 + m] = acc[v] + bv;
        }
    }
}

// ---------------------------------------------------------------------------
// Kernel 2: src[b,n,r] = <h[b,n,r,:], a[r,:C]> ; dst uses a[r,C:]
// ---------------------------------------------------------------------------
__global__ __launch_bounds__(256) void gat_attvec_kernel(
    const float* __restrict__ h, const float* __restrict__ a,
    float* __restrict__ srcv, float* __restrict__ dstv) {
    const int idx = blockIdx.x * 256 + threadIdx.x;   // < 512*512
    const int b = idx >> 9;
    const int p = idx & 511;          // n*4 + r
    const int n = p >> 2, r = p & 3;
    const float* hrow = h + ((size_t)b * NNODE + n) * HROW + r * COUT;
    const float* as = a + r * (2 * COUT);
    float s = 0.f, d = 0.f;
#pragma unroll 8
    for (int c = 0; c < COUT; ++c) {
        const float hv = hrow[c];
        s = fmaf(hv, as[c], s);
        d = fmaf(hv, as[COUT + c], d);
    }
    srcv[(size_t)b * JR + p] = s;
    dstv[(size_t)b * JR + p] = d;
}

// ---------------------------------------------------------------------------
// Kernel 3: fused mask -> joint softmax over (j,r) -> out = attn @ h
// grid = 512 (block per batch), block = 256 (8 waves, wave = 16-row i-tile).
// Pass 1: stream bonds once (float4 + global_prefetch), compute masked leaky
//         logits, online (max,sum), and park the 16x512 logit tile in LDS
//         (8 waves x 32 KB = 258 KB -- CDNA5's 320 KB WGP LDS).
// Pass 2: A-fragments = exp(logit - M)/S straight from ds_load_b64 (masked
//         entries hold -3.4e38 -> expf = 0); 128 K-steps x 4 C-tiles of
//         v_wmma_f32_16x16x4_f32.
// ---------------------------------------------------------------------------
__global__ __launch_bounds__(256) void gat_attn_out_kernel(
    const float* __restrict__ bonds, const float* __restrict__ h,
    const float* __restrict__ srcv, const float* __restrict__ dstv,
    float* __restrict__ out) {
    __shared__ float slg[8 * 16 * LROWS];   // 264192 B of the 320 KB WGP LDS

    const int b    = blockIdx.x;
    const int wave = threadIdx.x >> 5;
    const int lane = threadIdx.x & 31;
    const int m    = lane & 15;
    const int half = lane >> 4;
    const int i0   = wave * 16;
    const int i    = i0 + m;          // the row this lane serves in pass 1

    float* slgw = slg + wave * 16 * LROWS;          // this wave's logit tile

    const float* bondrow = bonds + ((size_t)b * NNODE + i) * JR;  // 512 contig
    const float* dstb    = dstv + (size_t)b * JR;
    const float* srow    = srcv + (size_t)b * JR + i * RREL;
    float sArr[4] = {srow[0], srow[1], srow[2], srow[3]};

    // ---- pass 1: masked logits -> LDS, online max/sum ---------------------
    float mx  = NEGBIG;
    float sum = 0.f;
    const int jr0 = half * (JR / 2);
    for (int t = 0; t < (JR / 2) / 4; ++t) {
        const int jr = jr0 + t * 4;               // j = jr>>2, r = 0..3
        __builtin_prefetch(bondrow + jr + 256, 0, 0);   // global_prefetch_b8
        const v4f b4 = *(const v4f*)(bondrow + jr);
        const v4f d4 = *(const v4f*)(dstb + jr);
        v4f lg4;
#pragma unroll
        for (int r = 0; r < 4; ++r) {
            float lg = sArr[r] + d4[r];
            lg = lg > 0.f ? lg : ALPHA * lg;
            const bool on  = (b4[r] == 1.0f);
            const float lgm = on ? lg : NEGBIG;
            lg4[r] = lgm;
            const float nm = fmaxf(mx, lgm);
            sum = sum * __expf(mx - nm) + (on ? __expf(lg - nm) : 0.f);
            mx  = nm;
        }
        *(v4f*)(slgw + m * LROWS + jr) = lg4;     // 16B-aligned ds_store_b128
    }
    // combine the two halves of each row (lane L <-> L^16), wave32 shuffle
    const float mx2  = __shfl_xor(mx, 16);
    const float sum2 = __shfl_xor(sum, 16);
    const float M    = fmaxf(mx, mx2);
    const float S    = sum * __expf(mx - M) + sum2 * __expf(mx2 - M);
    const float rinv = 1.0f / S;      // self-loop guarantees S > 0

    __syncthreads();                  // order LDS tile before fragment reads

    // ---- pass 2: attn @ h via WMMA ---------------------------------------
    // A slot v of this lane: row m, k = k0 + 2*half + v  -> ds_load_b64
    const float* hb = h + (size_t)b * NNODE * HROW;   // == (512 x 64) row-major
    const float* lrow = slgw + m * LROWS;

    v8f acc0 = {}, acc1 = {}, acc2 = {}, acc3 = {};
    for (int k0 = 0; k0 < JR; k0 += 4) {
        const v2f lg2 = *(const v2f*)(lrow + k0 + 2 * half);
        v2f afrag;
        afrag.x = __expf(lg2.x - M) * rinv;       // masked -> expf(-inf) = 0
        afrag.y = __expf(lg2.y - M) * rinv;

        const float* Bp = hb + (size_t)(k0 + 2 * half) * COUT + m;
        { v2f bf; bf.x = Bp[0];  bf.y = Bp[COUT + 0];  acc0 = wmma4(afrag, bf, acc0); }
        { v2f bf; bf.x = Bp[16]; bf.y = Bp[COUT + 16]; acc1 = wmma4(afrag, bf, acc1); }
        { v2f bf; bf.x = Bp[32]; bf.y = Bp[COUT + 32]; acc2 = wmma4(afrag, bf, acc2); }
        { v2f bf; bf.x = Bp[48]; bf.y = Bp[COUT + 48]; acc3 = wmma4(afrag, bf, acc3); }
    }

    // store D layout: row = i0 + v + 8*half, col = c0 + m
    float* ob = out + (size_t)b * NNODE * COUT;
#pragma unroll
    for (int v = 0; v < 8; ++v) {
        const size_t row = (size_t)(i0 + v + 8 * half) * COUT;
        ob[row +  0 + m] = acc0[v];
        ob[row + 16 + m] = acc1[v];
        ob[row + 32 + m] = acc2[v];
        ob[row + 48 + m] = acc3[v];
    }
}

// ---------------------------------------------------------------------------
extern "C" void kernel_launch(void* const* d_in, const int* in_sizes, int n_in,
                              void* d_out, int out_size, void* d_ws, size_t ws_size,
                              hipStream_t stream) {
    const float* atoms = (const float*)d_in[0];  // (512,128,64)
    const float* bonds = (const float*)d_in[1];  // (512,128,128,4)
    const float* W     = (const float*)d_in[2];  // (64,256)
    const float* bias  = (const float*)d_in[3];  // (256,)
    const float* a     = (const float*)d_in[4];  // (4,128)
    float* outp = (float*)d_out;                 // (512,128,64)

    float* h    = (float*)d_ws;                        // 512*128*256 f32 = 64 MiB
    float* srcv = h + (size_t)BATCH * NNODE * HROW;    // 512*512 f32
    float* dstv = srcv + (size_t)BATCH * JR;           // 512*512 f32

    gat_proj_kernel<<<BATCH, 256, 0, stream>>>(atoms, W, bias, h);
    gat_attvec_kernel<<<(BATCH * JR) / 256, 256, 0, stream>>>(h, a, srcv, dstv);
    gat_attn_out_kernel<<<BATCH, 256, 0, stream>>>(bonds, h, srcv, dstv, outp);
}